// GATNet_27857157882197
// MI455X (gfx1250) — compile-verified
//
#include <hip/hip_runtime.h>
#include <math.h>

typedef __attribute__((ext_vector_type(2))) float v2f;
typedef __attribute__((ext_vector_type(8))) float v8f;
typedef unsigned int u32;
typedef __attribute__((ext_vector_type(4))) u32 u32x4;
typedef __attribute__((ext_vector_type(8))) int i32x8;
typedef __attribute__((ext_vector_type(4))) int i32x4;

#define NEG_SLOPE 0.2f
#define KEY_NEG_INF 0x007FFFFFu  // fkey(-inf)

__device__ __forceinline__ unsigned int fkey(float f) {
  unsigned int b = __float_as_uint(f);
  unsigned int mask = (unsigned int)((int)b >> 31) | 0x80000000u;
  return b ^ mask;
}
__device__ __forceinline__ float funkey(unsigned int u) {
  unsigned int b = (u & 0x80000000u) ? (u ^ 0x80000000u) : ~u;
  return __uint_as_float(b);
}
__device__ __forceinline__ float lrelu(float v) {
  return v > 0.0f ? v : NEG_SLOPE * v;
}

__global__ __launch_bounds__(256) void fill_f32(float* __restrict__ p, long long n, float v) {
  long long i = (long long)blockIdx.x * blockDim.x + threadIdx.x;
  if (i < n) p[i] = v;
}
__global__ __launch_bounds__(256) void fill_u32(unsigned int* __restrict__ p, long long n, unsigned int v) {
  long long i = (long long)blockIdx.x * blockDim.x + threadIdx.x;
  if (i < n) p[i] = v;
}

// ---- TDM: async 2D tile load Global -> LDS (Tensor Data Mover, gfx1250) ----
// 2D descriptor: data_size=4B, tensor [tdim0 x tdim1] elems, tile [tile0 x tile1],
// row stride = stride0 elems. VADDR2/3 groups zero (<=2D tensor).
__device__ __forceinline__ void tdm_load_tile_2d(
    unsigned lds_off, const void* gptr,
    u32 tdim0, u32 tdim1, u32 tile0, u32 tile1, u32 stride0)
{
#if __has_builtin(__builtin_amdgcn_tensor_load_to_lds)
  unsigned long long ga = (unsigned long long)gptr;
  u32x4 g0;
  g0[0] = 1u;                                          // count=1, user descriptor
  g0[1] = lds_off;                                     // lds_addr (bytes)
  g0[2] = (u32)ga;                                     // global_addr[31:0]
  g0[3] = ((u32)(ga >> 32) & 0x01FFFFFFu) | (2u << 30); // global_addr[56:32] | type=2
  i32x8 g1;
  g1[0] = (int)(2u << 16);                             // data_size=4B; mask/flags=0
  g1[1] = (int)((tdim0 & 0xFFFFu) << 16);              // [63:48]=tensor_dim0 lo
  g1[2] = (int)((tdim0 >> 16) | ((tdim1 & 0xFFFFu) << 16)); // td0 hi | td1 lo
  g1[3] = (int)((tdim1 >> 16) | (tile0 << 16));        // td1 hi | tile_dim0
  g1[4] = (int)(tile1 & 0xFFFFu);                      // tile_dim1 | tile_dim2=0
  g1[5] = (int)stride0;                                // tensor_dim0_stride lo
  g1[6] = 0;                                           // stride0 hi | stride1 lo
  g1[7] = 0;
  i32x4 gz = {0, 0, 0, 0};
#if defined(__clang_major__) && (__clang_major__ >= 23)
  i32x8 gz8 = {0, 0, 0, 0, 0, 0, 0, 0};
  __builtin_amdgcn_tensor_load_to_lds(g0, g1, gz, gz, gz8, 0);
#else
  __builtin_amdgcn_tensor_load_to_lds(g0, g1, gz, gz, 0);
#endif
  __builtin_amdgcn_s_wait_tensorcnt(0);
#endif
}

// ---- Fused dual GEMM: Ol = A@Wl + bl, Or = A@Wr + br, via V_WMMA_F32_16X16X4_F32.
// One block per 16-row A tile (staged once into LDS by the TDM); wave w computes
// column tile nt=w for BOTH weight matrices, reusing its A fragments. K <= 128.
// blockDim.x = (NC/16) * 32.
__global__ __launch_bounds__(256) void wmma_dual_gemm_bias(
    const float* __restrict__ A,
    const float* __restrict__ Wl, const float* __restrict__ bl,
    const float* __restrict__ Wr, const float* __restrict__ br,
    float* __restrict__ Ol, float* __restrict__ Or,
    int K, int NC)
{
  __shared__ float shA[16 * 128];
  const int lane = threadIdx.x & 31;
  const int nt   = threadIdx.x >> 5;      // wave id == output column tile
  const int mt   = blockIdx.x;
  const float* __restrict__ Atile = A + (size_t)mt * 16 * K;

#if __has_builtin(__builtin_amdgcn_tensor_load_to_lds)
  if (nt == 0)                            // one TDM per block; TDM ignores EXEC
    tdm_load_tile_2d((unsigned)(size_t)(shA), Atile,
                     (u32)K, 16u, (u32)K, 16u, (u32)K);
#else
  for (int i = threadIdx.x; i < 16 * K; i += blockDim.x) shA[i] = Atile[i];
#endif
  __syncthreads();

  const int half = lane >> 4;             // 0: lanes 0-15 (K {0,1}); 1: lanes 16-31 (K {2,3})
  const int l16  = lane & 15;
  const float* __restrict__ sa  = shA + l16 * K + half * 2;
  const float* __restrict__ bcl = Wl + (size_t)(half * 2) * NC + (nt * 16 + l16);
  const float* __restrict__ bcr = Wr + (size_t)(half * 2) * NC + (nt * 16 + l16);

  v8f accl = {}, accr = {};
#pragma unroll 2
  for (int k = 0; k < K; k += 4) {
    v2f a, b0, b1;
    float2 av = *(const float2*)(sa + k);            // ds_load_b64 from LDS
    a.x = av.x; a.y = av.y;
    b0.x = bcl[(size_t)k * NC];
    b0.y = bcl[(size_t)(k + 1) * NC];
    b1.x = bcr[(size_t)k * NC];
    b1.y = bcr[(size_t)(k + 1) * NC];
    accl = __builtin_amdgcn_wmma_f32_16x16x4_f32(
        false, a, false, b0, (short)0, accl, false, false);
    accr = __builtin_amdgcn_wmma_f32_16x16x4_f32(
        false, a, false, b1, (short)0, accr, false, false);
  }

  const int col = nt * 16 + l16;
  const float bvl = bl[col];
  const float bvr = br[col];
  const int rowbase = mt * 16 + half * 8;  // C/D layout: vgpr v -> M=v (+8 for hi lanes)
#pragma unroll
  for (int v = 0; v < 8; ++v) {
    Ol[(size_t)(rowbase + v) * NC + col] = accl[v] + bvl;
    Or[(size_t)(rowbase + v) * NC + col] = accr[v] + bvr;
  }
}

// ---- Layer 1 edge score (H=4, C=32): one wave per edge; lane l handles channel l of each head.
__global__ __launch_bounds__(256) void score_max_h4(
    const long long* __restrict__ ei, long long E, long long Etot,
    const float* __restrict__ xl, const float* __restrict__ xr,
    const float* __restrict__ att, float* __restrict__ score,
    unsigned int* __restrict__ smaxkey)
{
  const int lane = threadIdx.x & 31;
  long long e = (long long)blockIdx.x * 8 + (threadIdx.x >> 5);
  if (e >= Etot) return;
  int src, dst;
  if (e < E) { src = (int)ei[e]; dst = (int)ei[E + e]; }
  else       { src = dst = (int)(e - E); }

  const float* __restrict__ xls = xl + (size_t)src * 128;
  const float* __restrict__ xrd = xr + (size_t)dst * 128;
  float s[4];
#pragma unroll
  for (int h = 0; h < 4; ++h)
    s[h] = att[h * 32 + lane] * lrelu(xls[h * 32 + lane] + xrd[h * 32 + lane]);
#pragma unroll
  for (int off = 16; off > 0; off >>= 1) {
#pragma unroll
    for (int h = 0; h < 4; ++h) s[h] += __shfl_xor(s[h], off, 32);
  }
  if (lane == 0) {
#pragma unroll
    for (int h = 0; h < 4; ++h) {
      score[e * 4 + h] = s[h];
      atomicMax(&smaxkey[(size_t)dst * 4 + h], fkey(s[h]));
    }
  }
}

// ---- Layer 2 edge score (H=1, C=64): one wave per edge; lane handles channels l, l+32.
__global__ __launch_bounds__(256) void score_max_h1(
    const long long* __restrict__ ei, long long E, long long Etot,
    const float* __restrict__ xl, const float* __restrict__ xr,
    const float* __restrict__ att, float* __restrict__ score,
    unsigned int* __restrict__ smaxkey)
{
  const int lane = threadIdx.x & 31;
  long long e = (long long)blockIdx.x * 8 + (threadIdx.x >> 5);
  if (e >= Etot) return;
  int src, dst;
  if (e < E) { src = (int)ei[e]; dst = (int)ei[E + e]; }
  else       { src = dst = (int)(e - E); }

  const float* __restrict__ xls = xl + (size_t)src * 64;
  const float* __restrict__ xrd = xr + (size_t)dst * 64;
  float s = att[lane]      * lrelu(xls[lane]      + xrd[lane])
          + att[lane + 32] * lrelu(xls[lane + 32] + xrd[lane + 32]);
#pragma unroll
  for (int off = 16; off > 0; off >>= 1) s += __shfl_xor(s, off, 32);
  if (lane == 0) {
    score[e] = s;
    atomicMax(&smaxkey[dst], fkey(s));
  }
}

// ---- exp(score - segmax) + segment-sum of denominators; 1 thread per (edge,head).
__global__ __launch_bounds__(256) void exp_sum(
    const long long* __restrict__ ei, long long E, long long Etot, int H,
    const unsigned int* __restrict__ smaxkey,
    float* __restrict__ score, float* __restrict__ denom)
{
  long long i = (long long)blockIdx.x * 256 + threadIdx.x;
  if (i >= Etot * H) return;
  long long e = i / H;
  int h = (int)(i - e * H);
  int dst = (e < E) ? (int)ei[E + e] : (int)(e - E);
  float m  = funkey(smaxkey[(size_t)dst * H + h]);
  float ex = __expf(score[i] - m);
  score[i] = ex;                               // reuse score buffer for exp values
  atomicAdd(&denom[(size_t)dst * H + h], ex);
}

// ---- Weighted scatter-add: agg[dst] += alpha * xl[src]. One wave per edge.
__global__ __launch_bounds__(256) void aggregate_h4(
    const long long* __restrict__ ei, long long E, long long Etot,
    const float* __restrict__ score, const float* __restrict__ denom,
    const float* __restrict__ xl, float* __restrict__ agg)
{
  const int lane = threadIdx.x & 31;
  long long e = (long long)blockIdx.x * 8 + (threadIdx.x >> 5);
  if (e >= Etot) return;
  int src, dst;
  if (e < E) { src = (int)ei[e]; dst = (int)ei[E + e]; }
  else       { src = dst = (int)(e - E); }
#pragma unroll
  for (int h = 0; h < 4; ++h) {
    float alpha = score[e * 4 + h] / denom[(size_t)dst * 4 + h];
    atomicAdd(&agg[(size_t)dst * 128 + h * 32 + lane],
              alpha * xl[(size_t)src * 128 + h * 32 + lane]);
  }
}

__global__ __launch_bounds__(256) void aggregate_h1(
    const long long* __restrict__ ei, long long E, long long Etot,
    const float* __restrict__ score, const float* __restrict__ denom,
    const float* __restrict__ xl, float* __restrict__ agg)
{
  const int lane = threadIdx.x & 31;
  long long e = (long long)blockIdx.x * 8 + (threadIdx.x >> 5);
  if (e >= Etot) return;
  int src, dst;
  if (e < E) { src = (int)ei[e]; dst = (int)ei[E + e]; }
  else       { src = dst = (int)(e - E); }
  float alpha = score[e] / denom[dst];
  atomicAdd(&agg[(size_t)dst * 64 + lane],      alpha * xl[(size_t)src * 64 + lane]);
  atomicAdd(&agg[(size_t)dst * 64 + lane + 32], alpha * xl[(size_t)src * 64 + lane + 32]);
}

__global__ __launch_bounds__(256) void bias_relu_inplace(
    float* __restrict__ h, const float* __restrict__ bias, long long n, int C)
{
  long long i = (long long)blockIdx.x * 256 + threadIdx.x;
  if (i >= n) return;
  float v = h[i] + bias[i % C];
  h[i] = v > 0.0f ? v : 0.0f;
}

__global__ __launch_bounds__(256) void bias_out(
    const float* __restrict__ agg, const float* __restrict__ bias,
    float* __restrict__ out, long long n, int C)
{
  long long i = (long long)blockIdx.x * 256 + threadIdx.x;
  if (i >= n) return;
  out[i] = agg[i] + bias[i % C];
}

static inline unsigned blk(long long n, int per) {
  return (unsigned)((n + per - 1) / per);
}

extern "C" void kernel_launch(void* const* d_in, const int* in_sizes, int n_in,
                              void* d_out, int out_size, void* d_ws, size_t ws_size,
                              hipStream_t stream) {
  const float*     x     = (const float*)d_in[0];
  const long long* ei    = (const long long*)d_in[1];
  const float* Wl1 = (const float*)d_in[2];  const float* bl1  = (const float*)d_in[3];
  const float* Wr1 = (const float*)d_in[4];  const float* br1  = (const float*)d_in[5];
  const float* att1= (const float*)d_in[6];  const float* bias1= (const float*)d_in[7];
  const float* Wl2 = (const float*)d_in[8];  const float* bl2  = (const float*)d_in[9];
  const float* Wr2 = (const float*)d_in[10]; const float* br2  = (const float*)d_in[11];
  const float* att2= (const float*)d_in[12]; const float* bias2= (const float*)d_in[13];

  const long long N    = in_sizes[0] / 128;
  const long long E    = in_sizes[1] / 2;
  const long long Etot = E + N;

  float* ws = (float*)d_ws;
  float* xl1   = ws; ws += N * 128;
  float* xr1   = ws; ws += N * 128;
  float* hbuf  = ws; ws += N * 128;   // layer-1 aggregation, then relu'd hidden
  float* xl2   = ws; ws += N * 64;
  float* xr2   = ws; ws += N * 64;
  float* agg2  = ws; ws += N * 64;
  unsigned int* smax1 = (unsigned int*)ws; ws += N * 4;
  float* denom1 = ws; ws += N * 4;
  unsigned int* smax2 = (unsigned int*)ws; ws += N;
  float* denom2 = ws; ws += N;
  float* score1 = ws; ws += Etot * 4;
  float* score2 = ws; ws += Etot;

  // init accumulators
  fill_u32<<<blk(N * 4, 256), 256, 0, stream>>>(smax1, N * 4, KEY_NEG_INF);
  fill_f32<<<blk(N * 4, 256), 256, 0, stream>>>(denom1, N * 4, 0.0f);
  fill_f32<<<blk(N * 128, 256), 256, 0, stream>>>(hbuf, N * 128, 0.0f);
  fill_u32<<<blk(N, 256), 256, 0, stream>>>(smax2, N, KEY_NEG_INF);
  fill_f32<<<blk(N, 256), 256, 0, stream>>>(denom2, N, 0.0f);
  fill_f32<<<blk(N * 64, 256), 256, 0, stream>>>(agg2, N * 64, 0.0f);

  // ---- layer 1: fused xl1/xr1 = x @ {Wl1,Wr1} + b (K=128, NC=128); 8 waves/block
  wmma_dual_gemm_bias<<<(unsigned)(N / 16), 256, 0, stream>>>(
      x, Wl1, bl1, Wr1, br1, xl1, xr1, 128, 128);

  score_max_h4<<<blk(Etot, 8), 256, 0, stream>>>(ei, E, Etot, xl1, xr1, att1, score1, smax1);
  exp_sum<<<blk(Etot * 4, 256), 256, 0, stream>>>(ei, E, Etot, 4, smax1, score1, denom1);
  aggregate_h4<<<blk(Etot, 8), 256, 0, stream>>>(ei, E, Etot, score1, denom1, xl1, hbuf);
  bias_relu_inplace<<<blk(N * 128, 256), 256, 0, stream>>>(hbuf, bias1, N * 128, 128);

  // ---- layer 2: fused xl2/xr2 = h @ {Wl2,Wr2} + b (K=128, NC=64); 4 waves/block
  wmma_dual_gemm_bias<<<(unsigned)(N / 16), 128, 0, stream>>>(
      hbuf, Wl2, bl2, Wr2, br2, xl2, xr2, 128, 64);

  score_max_h1<<<blk(Etot, 8), 256, 0, stream>>>(ei, E, Etot, xl2, xr2, att2, score2, smax2);
  exp_sum<<<blk(Etot, 256), 256, 0, stream>>>(ei, E, Etot, 1, smax2, score2, denom2);
  aggregate_h1<<<blk(Etot, 8), 256, 0, stream>>>(ei, E, Etot, score2, denom2, xl2, agg2);
  bias_out<<<blk(N * 64, 256), 256, 0, stream>>>(agg2, bias2, (float*)d_out, N * 64, 64);
}